// _FlatConvDrift_4088808866139
// MI455X (gfx1250) — compile-verified
//
#include <hip/hip_runtime.h>

typedef float v2f __attribute__((ext_vector_type(2)));
typedef float v8f __attribute__((ext_vector_type(8)));

#define N_NODES 100000
#define N_EDGES 50000
#define DIM     128
#define NNZ     1600000
#define MT_PER_BLOCK 4

// ---------------------------------------------------------------------------
// Zero-fill (harness poisons d_out/d_ws once; we must re-init accumulators
// deterministically every launch).
// ---------------------------------------------------------------------------
__global__ void zero_f32(float* __restrict__ p, int n) {
    int i = blockIdx.x * blockDim.x + threadIdx.x;
    if (i < n) p[i] = 0.0f;
}

// ---------------------------------------------------------------------------
// WMMA fp32 GEMM:  out[M x 128] = scale_rows(A[M x 128]) @ W[128 x 128] + bias
// Wave w of each block owns N-tile w (16 cols). B fragments (32 steps x v2f)
// are hoisted into registers; each wave sweeps MT_PER_BLOCK M-tiles.
// If USE_DEG, row r of A is scaled by 1/max(deg[r],1) (edge-mean fusion).
// All control flow around v_wmma is wave-uniform -> EXEC stays all-1s.
// ---------------------------------------------------------------------------
template <bool USE_DEG>
__global__ __launch_bounds__(256)
void gemm_wmma_bias(const float* __restrict__ A, const float* __restrict__ W,
                    const float* __restrict__ bias, float* __restrict__ out,
                    int M, const float* __restrict__ deg) {
    const int lane  = threadIdx.x & 31;
    const int wave  = threadIdx.x >> 5;      // 0..7  = N-tile index
    const int col   = lane & 15;             // column within 16-wide tile
    const int khalf = lane >> 4;             // 0 or 1: which K-pair this lane holds
    const int ncol  = wave * 16 + col;       // global output column 0..127

    // Hoist all B fragments: step kk covers K = 4*kk .. 4*kk+3.
    // Lane layout: lanes 0-15 hold K = 4kk+0,4kk+1 ; lanes 16-31 hold 4kk+2,4kk+3.
    v2f Bf[32];
#pragma unroll
    for (int kk = 0; kk < 32; ++kk) {
        const int k0 = kk * 4 + khalf * 2;
        Bf[kk].x = W[(k0 + 0) * DIM + ncol];
        Bf[kk].y = W[(k0 + 1) * DIM + ncol];
    }
    const float bv = bias[ncol];

    const int mbase = blockIdx.x * MT_PER_BLOCK;
#pragma unroll 1
    for (int mt = 0; mt < MT_PER_BLOCK; ++mt) {
        const int mtile = mbase + mt;
        if (mtile * 16 >= M) break;                 // wave-uniform guard
        const int rowA = mtile * 16 + col;          // A row this lane loads
        float dscale = 1.0f;
        if (USE_DEG) dscale = 1.0f / fmaxf(deg[rowA], 1.0f);

        v8f c = {};
#pragma unroll
        for (int kk = 0; kk < 32; ++kk) {
            const int k0 = kk * 4 + khalf * 2;
            v2f a;
            a.x = A[rowA * DIM + k0 + 0] * dscale;
            a.y = A[rowA * DIM + k0 + 1] * dscale;
            // D = A(16x4) x B(4x16) + C   -> v_wmma_f32_16x16x4_f32
            c = __builtin_amdgcn_wmma_f32_16x16x4_f32(
                    /*neg_a=*/false, a, /*neg_b=*/false, Bf[kk],
                    /*c_mod=*/(short)0, c, /*reuse_a=*/false, /*reuse_b=*/false);
        }

        // C/D layout: VGPR r -> M = mtile*16 + r + khalf*8, N = ncol.
#pragma unroll
        for (int r = 0; r < 8; ++r) {
            const int rowO = mtile * 16 + r + khalf * 8;
            out[rowO * DIM + ncol] = c[r] + bv;
        }
    }
}

// ---------------------------------------------------------------------------
// One wave per incidence pair: gather a 128-float row of src (float4/lane),
// scatter-add into accum row (L2-resident f32 atomics), count degree.
// ---------------------------------------------------------------------------
__global__ __launch_bounds__(256)
void scatter_rows(const float* __restrict__ src, const int* __restrict__ gidx,
                  const int* __restrict__ sidx, float* __restrict__ accum,
                  float* __restrict__ deg, int nnz) {
    const int pair = blockIdx.x * 8 + (threadIdx.x >> 5);
    if (pair >= nnz) return;
    const int lane = threadIdx.x & 31;
    const int g = gidx[pair];
    const int s = sidx[pair];
    const float4 v = *(const float4*)(src + (long long)g * DIM + lane * 4);
    float* dst = accum + (long long)s * DIM + lane * 4;
    atomicAdd(dst + 0, v.x);
    atomicAdd(dst + 1, v.y);
    atomicAdd(dst + 2, v.z);
    atomicAdd(dst + 3, v.w);
    if (lane == 0) atomicAdd(deg + s, 1.0f);
}

// ---------------------------------------------------------------------------
// out = relu(out / max(node_deg,1)) in place
// ---------------------------------------------------------------------------
__global__ void finalize_relu(float* __restrict__ out, const float* __restrict__ deg,
                              int total) {
    const int i = blockIdx.x * blockDim.x + threadIdx.x;
    if (i >= total) return;
    const int row = i >> 7;                    // / DIM
    const float v = out[i] / fmaxf(deg[row], 1.0f);
    out[i] = v > 0.0f ? v : 0.0f;
}

// ---------------------------------------------------------------------------
// Launch
// ---------------------------------------------------------------------------
extern "C" void kernel_launch(void* const* d_in, const int* in_sizes, int n_in,
                              void* d_out, int out_size, void* d_ws, size_t ws_size,
                              hipStream_t stream) {
    // inputs: 0=t, 1=y, 2=incidence[2,NNZ], 3=W1, 4=b1, 5=W2, 6=b2
    const float* y   = (const float*)d_in[1];
    const int*   inc = (const int*)d_in[2];
    const int*   n_idx = inc;            // incidence[0]
    const int*   e_idx = inc + NNZ;      // incidence[1]
    const float* W1 = (const float*)d_in[3];
    const float* b1 = (const float*)d_in[4];
    const float* W2 = (const float*)d_in[5];
    const float* b2 = (const float*)d_in[6];
    float* out = (float*)d_out;

    // workspace layout (floats)
    float* ws       = (float*)d_ws;
    float* msg      = ws;                                   // 100000*128
    float* edge_sum = msg + (size_t)N_NODES * DIM;          //  50000*128
    float* out_msg  = edge_sum + (size_t)N_EDGES * DIM;     //  50000*128
    float* edge_deg = out_msg + (size_t)N_EDGES * DIM;      //  50000
    float* node_deg = edge_deg + N_EDGES;                   // 100000

    const int threads = 256;
    // 1) zero accumulators (+ d_out, used as node_sum accumulator)
    zero_f32<<<(N_EDGES * DIM + threads - 1) / threads, threads, 0, stream>>>(edge_sum, N_EDGES * DIM);
    zero_f32<<<(N_EDGES + threads - 1) / threads, threads, 0, stream>>>(edge_deg, N_EDGES);
    zero_f32<<<(N_NODES + threads - 1) / threads, threads, 0, stream>>>(node_deg, N_NODES);
    zero_f32<<<(N_NODES * DIM + threads - 1) / threads, threads, 0, stream>>>(out, N_NODES * DIM);

    // 2) msg = X @ W1 + b1        (WMMA, M = 100000 -> 6250 M-tiles)
    {
        const int mtiles = N_NODES / 16;
        const int blocks = (mtiles + MT_PER_BLOCK - 1) / MT_PER_BLOCK;
        gemm_wmma_bias<false><<<blocks, threads, 0, stream>>>(y, W1, b1, msg, N_NODES, nullptr);
    }

    // 3) edge_sum += msg[n_idx] scattered by e_idx ; edge_deg counts
    scatter_rows<<<(NNZ + 7) / 8, threads, 0, stream>>>(msg, n_idx, e_idx, edge_sum, edge_deg, NNZ);

    // 4) out_msg = (edge_sum / max(edge_deg,1)) @ W2 + b2   (div fused into A load)
    {
        const int mtiles = N_EDGES / 16;
        const int blocks = (mtiles + MT_PER_BLOCK - 1) / MT_PER_BLOCK;
        gemm_wmma_bias<true><<<blocks, threads, 0, stream>>>(edge_sum, W2, b2, out_msg, N_EDGES, edge_deg);
    }

    // 5) d_out += out_msg[e_idx] scattered by n_idx ; node_deg counts
    scatter_rows<<<(NNZ + 7) / 8, threads, 0, stream>>>(out_msg, e_idx, n_idx, out, node_deg, NNZ);

    // 6) d_out = relu(d_out / max(node_deg,1))
    finalize_relu<<<(N_NODES * DIM + threads - 1) / threads, threads, 0, stream>>>(out, node_deg, N_NODES * DIM);
}